// RandomKSelection_20796231647785
// MI455X (gfx1250) — compile-verified
//
#include <hip/hip_runtime.h>
#include <stdint.h>

// Problem constants (from the reference's setup_inputs): B=16, C=64, H*W=65536, k=4096.
#define HW        65536u
#define K_SEL     4096u
#define THREADS   1024u

// Dynamic-LDS layout (bytes). No static __shared__ in the kernel, so the
// dynamic region starts at LDS offset 0 and these are the real hardware
// LDS byte addresses (needed for the async-copy inline asm).
#define W_OFF      0u                       // 65536 f32 weight row  (262144 B)
#define HIST_OFF   (HW * 4u)                // 256 u32 histogram     (  1024 B)
#define SEL_OFF    (HIST_OFF + 1024u)       // 4096 f32 selected x   ( 16384 B)
#define CTR_OFF    (SEL_OFF + K_SEL * 4u)   // 4 u32 counters/bcast  (    16 B)
#define SMEM_BYTES (CTR_OFF + 16u)          // 279568 B < 320 KB WGP LDS

extern "C" __global__ __launch_bounds__(THREADS)
void randk_select_sort(const float* __restrict__ x,
                       const float* __restrict__ wts,
                       float* __restrict__ out)
{
    extern __shared__ unsigned char smem[];
    unsigned* wbits = (unsigned*)(smem + W_OFF);
    unsigned* hist  = (unsigned*)(smem + HIST_OFF);
    float*    sel   = (float*)   (smem + SEL_OFF);
    unsigned* ctr   = (unsigned*)(smem + CTR_OFF);

    const unsigned tid = threadIdx.x;
    const unsigned row = blockIdx.x;
    const float* wrow = wts + (size_t)row * HW;
    const float* xrow = x   + (size_t)row * HW;
    float*       orow = out + (size_t)row * K_SEL;

    // ---- Stage the 256 KB weight row into LDS via CDNA5 async copies ----
    // 16384 chunks of 16 B; 16 chunks per thread, lane-consecutive chunks.
#pragma unroll
    for (unsigned t = 0; t < 16; ++t) {
        unsigned chunk = t * THREADS + tid;
        unsigned ldsa  = W_OFF + chunk * 16u;                       // LDS byte addr (VDST)
        unsigned long long ga =
            (unsigned long long)(const void*)(wrow + chunk * 4u);  // global addr (VADDR)
        asm volatile("global_load_async_to_lds_b128 %0, %1, off"
                     :: "v"(ldsa), "v"(ga) : "memory");
    }
    asm volatile("s_wait_asynccnt 0x0" ::: "memory");  // drain this wave's ASYNCcnt
    __syncthreads();                                   // all waves' chunks visible

    // ---- 4-pass MSB-first radix select: k-th largest weight bit pattern ----
    // Uniform [0,1) floats are positive => unsigned bit order == float order.
    unsigned prefix = 0, rem = K_SEL;
#pragma unroll 1
    for (int p = 0; p < 4; ++p) {
        const unsigned shift = (unsigned)(24 - 8 * p);
        if (tid < 256) hist[tid] = 0;
        __syncthreads();
        for (unsigned i = tid; i < HW; i += THREADS) {
            unsigned b = wbits[i];
            if (p == 0 || (b >> (shift + 8u)) == prefix)
                atomicAdd(&hist[(b >> shift) & 255u], 1u);
        }
        __syncthreads();
        if (tid == 0) {
            unsigned cum = 0, d = 0;
            for (int dd = 255; dd >= 0; --dd) {
                unsigned c = hist[dd];
                if (cum + c >= rem) { d = (unsigned)dd; break; }
                cum += c;
            }
            ctr[2] = d;          // chosen digit
            ctr[3] = rem - cum;  // elements == (prefix|d...) still needed
        }
        __syncthreads();
        prefix = (prefix << 8) | ctr[2];
        rem    = ctr[3];
        __syncthreads();
    }
    const unsigned T = prefix;         // k-th largest weight's bit pattern
    const unsigned G = K_SEL - rem;    // # weights strictly greater than T

    // ---- Select + gather the 4096 x values into LDS ----
    if (tid == 0) { ctr[0] = 0; ctr[1] = 0; }
    __syncthreads();
    for (unsigned i = tid; i < HW; i += THREADS) {
        unsigned b = wbits[i];
        if (b > T) {
            unsigned s = atomicAdd(&ctr[0], 1u);
            sel[s] = xrow[i];
        } else if (b == T) {
            unsigned e = atomicAdd(&ctr[1], 1u);
            if (e < rem) sel[G + e] = xrow[i];   // take exactly `rem` ties
        }
    }
    __syncthreads();

    // ---- Bitonic sort (ascending) of 4096 floats in LDS ----
    for (unsigned size = 2; size <= K_SEL; size <<= 1) {
        for (unsigned stride = size >> 1; stride > 0; stride >>= 1) {
            for (unsigned i = tid; i < K_SEL; i += THREADS) {
                unsigned j = i ^ stride;
                if (j > i) {                       // pair owner; pairs disjoint per stage
                    bool asc = ((i & size) == 0);
                    float a = sel[i], c = sel[j];
                    if ((a > c) == asc) { sel[i] = c; sel[j] = a; }
                }
            }
            __syncthreads();
        }
    }

    // ---- Coalesced store of the sorted row ----
    for (unsigned i = tid; i < K_SEL; i += THREADS) orow[i] = sel[i];
}

extern "C" void kernel_launch(void* const* d_in, const int* in_sizes, int n_in,
                              void* d_out, int out_size, void* d_ws, size_t ws_size,
                              hipStream_t stream) {
    const float* x = (const float*)d_in[0];   // [B, C, H, W] f32
    const float* w = (const float*)d_in[1];   // [B, C, H*W]  f32
    float* out = (float*)d_out;               // [B, C, 4096] f32
    (void)d_ws; (void)ws_size; (void)n_in; (void)out_size;

    const int rows = in_sizes[1] / (int)HW;   // B*C = 1024 independent rows
    randk_select_sort<<<dim3(rows), dim3(THREADS), SMEM_BYTES, stream>>>(x, w, out);
}